// Crf_6451040878896
// MI455X (gfx1250) — compile-verified
//
#include <hip/hip_runtime.h>

typedef float v2f __attribute__((ext_vector_type(2)));
typedef float v8f __attribute__((ext_vector_type(8)));

namespace {
constexpr int Bv = 64;     // batch
constexpr int Tv = 1024;   // seq len
constexpr int Hv = 1024;   // hidden
constexpr int Nv = 64;     // num tags
}

// ---------------------------------------------------------------------------
// Kernel 1: emission[M,N] = X[M,H] * W[N,H]^T + bias   (M = B*T)
// f32 WMMA 16x16x4. One wave -> 16 rows x 64 cols. Block = 8 waves = 128 rows.
//
// A (16x4 f32): lane l holds row (l&15); VGPR0/1 hold K = 2*(l>>4), 2*(l>>4)+1.
// B (4x16 f32): lane l holds col (l&15); VGPR0/1 hold K = 2*(l>>4), 2*(l>>4)+1.
// C (16x16 f32): VGPR r, lanes 0-15 -> M=r, lanes 16-31 -> M=r+8; col = l&15.
// ---------------------------------------------------------------------------
__global__ __launch_bounds__(256) void emission_gemm(
    const float* __restrict__ X, const float* __restrict__ W,
    const float* __restrict__ bias, float* __restrict__ out)
{
    const int lane  = threadIdx.x & 31;
    const int wave  = threadIdx.x >> 5;
    const int mbase = (blockIdx.x * 8 + wave) * 16;
    const int r16   = lane & 15;     // A-row / B-col / C-col within 16-tile
    const int kh    = lane >> 4;     // K-half selector

    const float* xrow = X + (size_t)(mbase + r16) * Hv + 2 * kh;
    const float* w0   = W + (size_t)(r16) * Hv + 2 * kh;        // cols  0..15
    const float* w1   = w0 + (size_t)16 * Hv;                   // cols 16..31
    const float* w2   = w0 + (size_t)32 * Hv;                   // cols 32..47
    const float* w3   = w0 + (size_t)48 * Hv;                   // cols 48..63

    v8f acc0 = {}, acc1 = {}, acc2 = {}, acc3 = {};

    #pragma unroll 4
    for (int k = 0; k < Hv; k += 4) {
        v2f a  = *(const v2f*)(xrow + k);
        v2f b0 = *(const v2f*)(w0 + k);
        v2f b1 = *(const v2f*)(w1 + k);
        v2f b2 = *(const v2f*)(w2 + k);
        v2f b3 = *(const v2f*)(w3 + k);
        acc0 = __builtin_amdgcn_wmma_f32_16x16x4_f32(false, a, false, b0,
                                                     (short)0, acc0, false, false);
        acc1 = __builtin_amdgcn_wmma_f32_16x16x4_f32(false, a, false, b1,
                                                     (short)0, acc1, false, false);
        acc2 = __builtin_amdgcn_wmma_f32_16x16x4_f32(false, a, false, b2,
                                                     (short)0, acc2, false, false);
        acc3 = __builtin_amdgcn_wmma_f32_16x16x4_f32(false, a, false, b3,
                                                     (short)0, acc3, false, false);
    }

    const int rowoff = mbase + kh * 8;   // first of this lane-half's 8 rows

    {
        float bv = bias[0 * 16 + r16];
        float* o = out + (size_t)rowoff * Nv + 0 * 16 + r16;
        #pragma unroll
        for (int r = 0; r < 8; ++r) o[(size_t)r * Nv] = acc0[r] + bv;
    }
    {
        float bv = bias[1 * 16 + r16];
        float* o = out + (size_t)rowoff * Nv + 1 * 16 + r16;
        #pragma unroll
        for (int r = 0; r < 8; ++r) o[(size_t)r * Nv] = acc1[r] + bv;
    }
    {
        float bv = bias[2 * 16 + r16];
        float* o = out + (size_t)rowoff * Nv + 2 * 16 + r16;
        #pragma unroll
        for (int r = 0; r < 8; ++r) o[(size_t)r * Nv] = acc2[r] + bv;
    }
    {
        float bv = bias[3 * 16 + r16];
        float* o = out + (size_t)rowoff * Nv + 3 * 16 + r16;
        #pragma unroll
        for (int r = 0; r < 8; ++r) o[(size_t)r * Nv] = acc3[r] + bv;
    }
}

// ---------------------------------------------------------------------------
// Kernel 2: Viterbi forward. One block per batch, one thread per tag (N=64).
// trans column held in registers; score exchanged via 256B of LDS.
// History: u8 argmax per (t,j); identity index stored when masked off so the
// backtrack pass needs no mask.
// ---------------------------------------------------------------------------
__global__ __launch_bounds__(64) void viterbi_fwd(
    const float* __restrict__ emission, const unsigned char* __restrict__ mask,
    const float* __restrict__ start_trans, const float* __restrict__ end_trans,
    const float* __restrict__ trans, unsigned char* __restrict__ hist,
    int* __restrict__ last_tag, float* __restrict__ pred)
{
    __shared__ float s_score[Nv];
    const int b = blockIdx.x;
    const int j = threadIdx.x;

    float tcol[Nv];
    #pragma unroll
    for (int i = 0; i < Nv; ++i) tcol[i] = trans[i * Nv + j];

    const float*         eb = emission + (size_t)b * Tv * Nv;
    const unsigned char* mb = mask + (size_t)b * Tv;
    unsigned char*       hb = hist + (size_t)b * (Tv - 1) * Nv;

    float score = start_trans[j] + eb[j];
    s_score[j] = score;
    __syncthreads();

    for (int t = 1; t < Tv; ++t) {
        float e_t = eb[(size_t)t * Nv + j];
        unsigned int m = mb[t];

        float best = s_score[0] + tcol[0];
        int   arg  = 0;
        #pragma unroll
        for (int i = 1; i < Nv; ++i) {
            float c = s_score[i] + tcol[i];
            if (c > best) { best = c; arg = i; }   // strict '>' == first argmax
        }
        __syncthreads();   // all lanes finished reading s_score

        hb[(size_t)(t - 1) * Nv + j] = (unsigned char)(m ? arg : j);
        float ns = m ? (best + e_t) : score;
        s_score[j] = ns;
        score = ns;
        __syncthreads();
    }

    s_score[j] = score + end_trans[j];
    __syncthreads();
    if (j == 0) {
        float best = s_score[0];
        int   arg  = 0;
        for (int i = 1; i < Nv; ++i) {
            float c = s_score[i];
            if (c > best) { best = c; arg = i; }
        }
        last_tag[b] = arg;
        pred[(size_t)b * Tv + (Tv - 1)] = (float)arg;
    }
}

// ---------------------------------------------------------------------------
// Kernel 3: backtrack. One block per batch; stage 64KB history into LDS,
// then thread 0 walks the dependent chain in LDS instead of HBM.
// ---------------------------------------------------------------------------
__global__ __launch_bounds__(256) void viterbi_back(
    const unsigned char* __restrict__ hist, const int* __restrict__ last_tag,
    float* __restrict__ pred)
{
    __shared__ unsigned char s_hist[(Tv - 1) * Nv];   // 65472 bytes
    const int b = blockIdx.x;

    const uint4* src = (const uint4*)(hist + (size_t)b * (Tv - 1) * Nv);
    uint4*       dst = (uint4*)s_hist;
    const int n16 = (Tv - 1) * Nv / 16;
    for (int i = threadIdx.x; i < n16; i += blockDim.x) dst[i] = src[i];
    __syncthreads();

    if (threadIdx.x == 0) {
        int tag = last_tag[b];
        float* p = pred + (size_t)b * Tv;
        for (int t = Tv - 2; t >= 0; --t) {
            tag = s_hist[t * Nv + tag];   // masked steps stored identity
            p[t] = (float)tag;
        }
    }
}

// ---------------------------------------------------------------------------
extern "C" void kernel_launch(void* const* d_in, const int* in_sizes, int n_in,
                              void* d_out, int out_size, void* d_ws, size_t ws_size,
                              hipStream_t stream)
{
    (void)in_sizes; (void)n_in; (void)out_size; (void)ws_size;

    const float*         text  = (const float*)d_in[0];          // [B,T,H]
    const unsigned char* mask  = (const unsigned char*)d_in[1];  // [B,T] bool
    const float*         W_em  = (const float*)d_in[2];          // [N,H]
    const float*         b_em  = (const float*)d_in[3];          // [N]
    const float*         strt  = (const float*)d_in[4];          // [N]
    const float*         endt  = (const float*)d_in[5];          // [N]
    const float*         trans = (const float*)d_in[6];          // [N,N]

    float* out      = (float*)d_out;
    float* emission = out;                               // B*T*N f32
    float* pred     = out + (size_t)Bv * Tv * Nv;        // B*T tags (as f32)

    unsigned char* hist = (unsigned char*)d_ws;                          // B*(T-1)*N u8
    int* last = (int*)((char*)d_ws + (size_t)Bv * (Tv - 1) * Nv);        // B ints

    const int M = Bv * Tv;                  // 65536 rows
    emission_gemm<<<M / 128, 256, 0, stream>>>(text, W_em, b_em, emission);
    viterbi_fwd<<<Bv, 64, 0, stream>>>(emission, mask, strt, endt, trans,
                                       hist, last, pred);
    viterbi_back<<<Bv, 256, 0, stream>>>(hist, last, pred);
}